// POELayer_66322884985440
// MI455X (gfx1250) — compile-verified
//
#include <hip/hip_runtime.h>
#include <cstdint>

// ---------------------------------------------------------------------------
// POE forward-kinematics chain, MI455X (gfx1250, wave32).
//   kernel 1 (1 wave): per-joint wave-uniform constants + tool matrices -> d_ws
//   kernel 2: one thread per batch element; q staged into LDS via CDNA5
//             async global->LDS b128 copies (ASYNCcnt); chain in registers,
//             per-joint constants via uniform (scalar) loads; HW sincos.
// Roofline: ~226 MB traffic -> ~10us floor; trimmed VALU to match.
// ---------------------------------------------------------------------------

#define JNT 12
#define BLK 256
#define EPSF 1e-12f

#define REC_F   24                 // floats per (joint,type) record
#define TOOL_F  12                 // floats per tool matrix (R rowmajor + p)
#define CST_REC_FLOATS (24 * REC_F)            // 576
#define CST_TOTAL_FLOATS (CST_REC_FLOATS + 3 * TOOL_F)  // 612
#define CST_BYTES (CST_TOTAL_FLOATS * 4)

// ---- CDNA5 async global->LDS copy (tracked by ASYNCcnt) -------------------
__device__ __forceinline__ void async_copy_b128(unsigned lds_off, const void* gptr) {
    asm volatile("global_load_async_to_lds_b128 %0, %1, off"
                 :: "v"(lds_off), "v"(gptr)
                 : "memory");
}
__device__ __forceinline__ void wait_async_zero() {
    asm volatile("s_wait_asynccnt 0" ::: "memory");
}
__device__ __forceinline__ unsigned lds_byte_addr(const void* p) {
    return (unsigned)(uintptr_t)p;   // low 32 bits of generic LDS ptr = LDS offset
}

struct Frame {
    float r00, r01, r02;
    float r10, r11, r12;
    float r20, r21, r22;
    float p0, p1, p2;
};

__device__ __forceinline__ void store6(float* __restrict__ d,
                                       float a, float b, float c,
                                       float e, float f, float g) {
    *(float2*)(d + 0) = make_float2(a, b);   // (b*72 + j*6) even -> 8B aligned
    *(float2*)(d + 2) = make_float2(c, e);
    *(float2*)(d + 4) = make_float2(f, g);
}

// ===========================================================================
// Kernel 1: wave-uniform precompute (runs once per call, 1 wave).
// Record layout (24 floats):
//  [0..2] w  [3..5] v  [6..8] w x v  [9..11] (w.v)w - ww v
//  [12] a  [13] 1/a  [14] 1/a^2  [15] 1/a^3
//  [16..18] wi^2 - ww (diag)  [19..21] wxwy, wxwz, wywz  [22..23] pad
// ===========================================================================
extern "C" __global__ __launch_bounds__(32)
void poe_precompute(const float* __restrict__ rev,
                    const float* __restrict__ pri,
                    const float* __restrict__ pos,
                    const float* __restrict__ ori,
                    float* __restrict__ cst) {
    const int t = threadIdx.x;
    if (t < 24) {
        const int j = t >> 1;
        const float* xi = (t & 1) ? (pri + j * 6) : (rev + j * 6);
        float wx = xi[0], wy = xi[1], wz = xi[2];
        float vx = xi[3], vy = xi[4], vz = xi[5];
        float ww   = wx * wx + wy * wy + wz * wz;
        float a    = sqrtf(ww + EPSF);
        float inva = 1.0f / a;
        float cx = wy * vz - wz * vy;
        float cy = wz * vx - wx * vz;
        float cz = wx * vy - wy * vx;
        float wv = wx * vx + wy * vy + wz * vz;
        float* r = cst + t * REC_F;
        r[0] = wx;  r[1] = wy;  r[2] = wz;
        r[3] = vx;  r[4] = vy;  r[5] = vz;
        r[6] = cx;  r[7] = cy;  r[8] = cz;
        r[9]  = wv * wx - ww * vx;
        r[10] = wv * wy - ww * vy;
        r[11] = wv * wz - ww * vz;
        r[12] = a;  r[13] = inva;  r[14] = inva * inva;  r[15] = inva * inva * inva;
        r[16] = wx * wx - ww;  r[17] = wy * wy - ww;  r[18] = wz * wz - ww;
        r[19] = wx * wy;       r[20] = wx * wz;       r[21] = wy * wz;
        r[22] = 0.f; r[23] = 0.f;
    } else if (t < 27) {
        const int k = t - 24;
        const int j = 4 * k + 3;                 // BRANCH_IDX = {3,7,11}
        const float* o = ori + j * 3;
        float cr = cosf(o[0]), sr = sinf(o[0]);  // accurate libm (runs once)
        float cp = cosf(o[1]), sp = sinf(o[1]);
        float cy = cosf(o[2]), sy = sinf(o[2]);
        float* r = cst + CST_REC_FLOATS + k * TOOL_F;
        r[0] = cy * cp;  r[1] = cy * sp * sr - sy * cr;  r[2] = cy * sp * cr + sy * sr;
        r[3] = sy * cp;  r[4] = sy * sp * sr + cy * cr;  r[5] = sy * sp * cr - cy * sr;
        r[6] = -sp;      r[7] = cp * sr;                 r[8] = cp * cr;
        r[9] = pos[j * 3 + 0]; r[10] = pos[j * 3 + 1]; r[11] = pos[j * 3 + 2];
    }
}

// ===========================================================================
// Half-step using the precomputed record (all record values wave-uniform).
// ===========================================================================
__device__ __forceinline__ void half_step_c(Frame& F,
                                            const float* __restrict__ c,
                                            float th,
                                            float* __restrict__ dstT) {
    const float wx = c[0], wy = c[1], wz = c[2];
    const float vx = c[3], vy = c[4], vz = c[5];

    // ---- adj_inv_apply(out, xi): wp = R^T w ; vp = R^T (v - p x w) ----
    float tx = vx - (F.p1 * wz - F.p2 * wy);
    float ty = vy - (F.p2 * wx - F.p0 * wz);
    float tz = vz - (F.p0 * wy - F.p1 * wx);
    float wpx = F.r00 * wx + F.r10 * wy + F.r20 * wz;
    float wpy = F.r01 * wx + F.r11 * wy + F.r21 * wz;
    float wpz = F.r02 * wx + F.r12 * wy + F.r22 * wz;
    float vpx = F.r00 * tx + F.r10 * ty + F.r20 * tz;
    float vpy = F.r01 * tx + F.r11 * ty + F.r21 * tz;
    float vpz = F.r02 * tx + F.r12 * ty + F.r22 * tz;
    store6(dstT, wpx, wpy, wpz, vpx, vpy, vpz);

    // ---- exp(hat(xi)*th), analytic with precomputed uniforms ----
    float ang = c[12] * th;           // a * th
    float s, co;
    __sincosf(ang, &s, &co);          // hw v_sin/v_cos (args are small)
    float k1 = s * c[13];             // sin/a
    float k2 = (1.0f - co) * c[14];   // (1-cos)/a^2
    float k3 = (ang - s) * c[15];     // (ang-sin)/a^3

    float k1x = k1 * wx, k1y = k1 * wy, k1z = k1 * wz;
    float k2xy = k2 * c[19], k2xz = k2 * c[20], k2yz = k2 * c[21];
    float d00 = 1.0f + k2 * c[16];
    float d11 = 1.0f + k2 * c[17];
    float d22 = 1.0f + k2 * c[18];
    float d01 = k2xy - k1z, d10 = k2xy + k1z;
    float d02 = k2xz + k1y, d20 = k2xz - k1y;
    float d12 = k2yz - k1x, d21 = k2yz + k1x;

    float pdx = th * vx + k2 * c[6] + k3 * c[9];
    float pdy = th * vy + k2 * c[7] + k3 * c[10];
    float pdz = th * vz + k2 * c[8] + k3 * c[11];

    // ---- out = out @ [D pd; 0 1] ----
    float n00 = F.r00 * d00 + F.r01 * d10 + F.r02 * d20;
    float n01 = F.r00 * d01 + F.r01 * d11 + F.r02 * d21;
    float n02 = F.r00 * d02 + F.r01 * d12 + F.r02 * d22;
    float n10 = F.r10 * d00 + F.r11 * d10 + F.r12 * d20;
    float n11 = F.r10 * d01 + F.r11 * d11 + F.r12 * d21;
    float n12 = F.r10 * d02 + F.r11 * d12 + F.r12 * d22;
    float n20 = F.r20 * d00 + F.r21 * d10 + F.r22 * d20;
    float n21 = F.r20 * d01 + F.r21 * d11 + F.r22 * d21;
    float n22 = F.r20 * d02 + F.r21 * d12 + F.r22 * d22;
    float np0 = F.r00 * pdx + F.r01 * pdy + F.r02 * pdz + F.p0;
    float np1 = F.r10 * pdx + F.r11 * pdy + F.r12 * pdz + F.p1;
    float np2 = F.r20 * pdx + F.r21 * pdy + F.r22 * pdz + F.p2;
    F.r00 = n00; F.r01 = n01; F.r02 = n02;
    F.r10 = n10; F.r11 = n11; F.r12 = n12;
    F.r20 = n20; F.r21 = n21; F.r22 = n22;
    F.p0 = np0; F.p1 = np1; F.p2 = np2;
}

// ===========================================================================
// Kernel 2: main chain.
// ===========================================================================
extern "C" __global__ __launch_bounds__(BLK)
void poe_fk_kernel(const float* __restrict__ q,     // (B,12,2)
                   const float* __restrict__ cst,   // precomputed table
                   float* __restrict__ outs,        // (B,3,4,4)
                   float* __restrict__ revls,       // (B,12,6)
                   float* __restrict__ prils,       // (B,12,6)
                   int B) {
    __shared__ float lq[BLK * 24];                  // 24 KB q slab

    const int tid = threadIdx.x;
    const int b   = blockIdx.x * BLK + tid;
    const bool full_block = (blockIdx.x * BLK + BLK) <= B;

    if (full_block) {
        const char* gbase = (const char*)(q + (size_t)blockIdx.x * BLK * 24);
        unsigned lbase = lds_byte_addr(&lq[0]);
#pragma unroll
        for (int i = 0; i < 6; ++i) {
            unsigned off = (unsigned)((tid + i * BLK) * 16);
            async_copy_b128(lbase + off, gbase + off);
        }
    } else if (b < B) {
#pragma unroll
        for (int k = 0; k < 24; ++k) lq[tid * 24 + k] = q[(size_t)b * 24 + k];
    }
    wait_async_zero();
    __syncthreads();
    if (b >= B) return;

    Frame F = {1.f, 0.f, 0.f, 0.f, 1.f, 0.f, 0.f, 0.f, 1.f, 0.f, 0.f, 0.f};
    const int    qbase = tid * 24;
    const size_t b72   = (size_t)b * 72;
    const size_t b48   = (size_t)b * 48;

#pragma unroll
    for (int j = 0; j < JNT; ++j) {
        float2 qq = *(const float2*)&lq[qbase + 2 * j];

        half_step_c(F, cst + (2 * j + 0) * REC_F, qq.x, revls + b72 + j * 6);
        half_step_c(F, cst + (2 * j + 1) * REC_F, qq.y, prils + b72 + j * 6);

        if ((j & 3) == 3) {
            const int k = j >> 2;
            const float* tl = cst + CST_REC_FLOATS + k * TOOL_F;
            float t00 = tl[0], t01 = tl[1], t02 = tl[2];
            float t10 = tl[3], t11 = tl[4], t12 = tl[5];
            float t20 = tl[6], t21 = tl[7], t22 = tl[8];
            float tp0 = tl[9], tp1 = tl[10], tp2 = tl[11];

            float m00 = F.r00 * t00 + F.r01 * t10 + F.r02 * t20;
            float m01 = F.r00 * t01 + F.r01 * t11 + F.r02 * t21;
            float m02 = F.r00 * t02 + F.r01 * t12 + F.r02 * t22;
            float m10 = F.r10 * t00 + F.r11 * t10 + F.r12 * t20;
            float m11 = F.r10 * t01 + F.r11 * t11 + F.r12 * t21;
            float m12 = F.r10 * t02 + F.r11 * t12 + F.r12 * t22;
            float m20 = F.r20 * t00 + F.r21 * t10 + F.r22 * t20;
            float m21 = F.r20 * t01 + F.r21 * t11 + F.r22 * t21;
            float m22 = F.r20 * t02 + F.r21 * t12 + F.r22 * t22;
            float mp0 = F.r00 * tp0 + F.r01 * tp1 + F.r02 * tp2 + F.p0;
            float mp1 = F.r10 * tp0 + F.r11 * tp1 + F.r12 * tp2 + F.p1;
            float mp2 = F.r20 * tp0 + F.r21 * tp1 + F.r22 * tp2 + F.p2;

            float4* op = (float4*)(outs + b48 + (size_t)k * 16);
            op[0] = make_float4(m00, m01, m02, mp0);
            op[1] = make_float4(m10, m11, m12, mp1);
            op[2] = make_float4(m20, m21, m22, mp2);
            op[3] = make_float4(0.f, 0.f, 0.f, 1.f);
        }
    }
}

// ===========================================================================
// Fallback (self-contained) in case ws_size is too small: computes the
// per-joint uniforms inline (accurate libm trig).
// ===========================================================================
__device__ __forceinline__ void half_step_s(Frame& F,
                                            const float* __restrict__ xi,
                                            float th,
                                            float* __restrict__ dstT) {
    float wx = xi[0], wy = xi[1], wz = xi[2];
    float vx = xi[3], vy = xi[4], vz = xi[5];
    float tx = vx - (F.p1 * wz - F.p2 * wy);
    float ty = vy - (F.p2 * wx - F.p0 * wz);
    float tz = vz - (F.p0 * wy - F.p1 * wx);
    float wpx = F.r00 * wx + F.r10 * wy + F.r20 * wz;
    float wpy = F.r01 * wx + F.r11 * wy + F.r21 * wz;
    float wpz = F.r02 * wx + F.r12 * wy + F.r22 * wz;
    float vpx = F.r00 * tx + F.r10 * ty + F.r20 * tz;
    float vpy = F.r01 * tx + F.r11 * ty + F.r21 * tz;
    float vpz = F.r02 * tx + F.r12 * ty + F.r22 * tz;
    store6(dstT, wpx, wpy, wpz, vpx, vpy, vpz);

    float ww = wx * wx + wy * wy + wz * wz;
    float a = sqrtf(ww + EPSF);
    float inva = 1.0f / a, inva2 = inva * inva, inva3 = inva2 * inva;
    float ang = a * th;
    float s = sinf(ang), co = cosf(ang);
    float k1 = s * inva, k2 = (1.0f - co) * inva2, k3 = (ang - s) * inva3;
    float d00 = 1.0f + k2 * (wx * wx - ww);
    float d01 = k1 * (-wz) + k2 * (wx * wy);
    float d02 = k1 * ( wy) + k2 * (wx * wz);
    float d10 = k1 * ( wz) + k2 * (wy * wx);
    float d11 = 1.0f + k2 * (wy * wy - ww);
    float d12 = k1 * (-wx) + k2 * (wy * wz);
    float d20 = k1 * (-wy) + k2 * (wz * wx);
    float d21 = k1 * ( wx) + k2 * (wz * wy);
    float d22 = 1.0f + k2 * (wz * wz - ww);
    float cwx = wy * vz - wz * vy;
    float cwy = wz * vx - wx * vz;
    float cwz = wx * vy - wy * vx;
    float wv = wx * vx + wy * vy + wz * vz;
    float pdx = th * vx + k2 * cwx + k3 * (wv * wx - ww * vx);
    float pdy = th * vy + k2 * cwy + k3 * (wv * wy - ww * vy);
    float pdz = th * vz + k2 * cwz + k3 * (wv * wz - ww * vz);

    float n00 = F.r00 * d00 + F.r01 * d10 + F.r02 * d20;
    float n01 = F.r00 * d01 + F.r01 * d11 + F.r02 * d21;
    float n02 = F.r00 * d02 + F.r01 * d12 + F.r02 * d22;
    float n10 = F.r10 * d00 + F.r11 * d10 + F.r12 * d20;
    float n11 = F.r10 * d01 + F.r11 * d11 + F.r12 * d21;
    float n12 = F.r10 * d02 + F.r11 * d12 + F.r12 * d22;
    float n20 = F.r20 * d00 + F.r21 * d10 + F.r22 * d20;
    float n21 = F.r20 * d01 + F.r21 * d11 + F.r22 * d21;
    float n22 = F.r20 * d02 + F.r21 * d12 + F.r22 * d22;
    float np0 = F.r00 * pdx + F.r01 * pdy + F.r02 * pdz + F.p0;
    float np1 = F.r10 * pdx + F.r11 * pdy + F.r12 * pdz + F.p1;
    float np2 = F.r20 * pdx + F.r21 * pdy + F.r22 * pdz + F.p2;
    F.r00 = n00; F.r01 = n01; F.r02 = n02;
    F.r10 = n10; F.r11 = n11; F.r12 = n12;
    F.r20 = n20; F.r21 = n21; F.r22 = n22;
    F.p0 = np0; F.p1 = np1; F.p2 = np2;
}

extern "C" __global__ __launch_bounds__(BLK)
void poe_fk_fallback(const float* __restrict__ q,
                     const float* __restrict__ rev,
                     const float* __restrict__ pri,
                     const float* __restrict__ pos,
                     const float* __restrict__ ori,
                     float* __restrict__ outs,
                     float* __restrict__ revls,
                     float* __restrict__ prils,
                     int B) {
    __shared__ float lq[BLK * 24];
    const int tid = threadIdx.x;
    const int b   = blockIdx.x * BLK + tid;
    const bool full_block = (blockIdx.x * BLK + BLK) <= B;
    if (full_block) {
        const char* gbase = (const char*)(q + (size_t)blockIdx.x * BLK * 24);
        unsigned lbase = lds_byte_addr(&lq[0]);
#pragma unroll
        for (int i = 0; i < 6; ++i) {
            unsigned off = (unsigned)((tid + i * BLK) * 16);
            async_copy_b128(lbase + off, gbase + off);
        }
    } else if (b < B) {
#pragma unroll
        for (int k = 0; k < 24; ++k) lq[tid * 24 + k] = q[(size_t)b * 24 + k];
    }
    wait_async_zero();
    __syncthreads();
    if (b >= B) return;

    Frame F = {1.f, 0.f, 0.f, 0.f, 1.f, 0.f, 0.f, 0.f, 1.f, 0.f, 0.f, 0.f};
    const int    qbase = tid * 24;
    const size_t b72   = (size_t)b * 72;
    const size_t b48   = (size_t)b * 48;
    for (int j = 0; j < JNT; ++j) {
        float2 qq = *(const float2*)&lq[qbase + 2 * j];
        half_step_s(F, rev + j * 6, qq.x, revls + b72 + j * 6);
        half_step_s(F, pri + j * 6, qq.y, prils + b72 + j * 6);
        if ((j & 3) == 3) {
            const int k = j >> 2;
            const float* o = ori + j * 3;
            float cr = cosf(o[0]), sr = sinf(o[0]);
            float cp = cosf(o[1]), sp = sinf(o[1]);
            float cy = cosf(o[2]), sy = sinf(o[2]);
            float t00 = cy * cp, t01 = cy * sp * sr - sy * cr, t02 = cy * sp * cr + sy * sr;
            float t10 = sy * cp, t11 = sy * sp * sr + cy * cr, t12 = sy * sp * cr - cy * sr;
            float t20 = -sp,     t21 = cp * sr,                t22 = cp * cr;
            float tp0 = pos[j * 3 + 0], tp1 = pos[j * 3 + 1], tp2 = pos[j * 3 + 2];
            float m00 = F.r00 * t00 + F.r01 * t10 + F.r02 * t20;
            float m01 = F.r00 * t01 + F.r01 * t11 + F.r02 * t21;
            float m02 = F.r00 * t02 + F.r01 * t12 + F.r02 * t22;
            float m10 = F.r10 * t00 + F.r11 * t10 + F.r12 * t20;
            float m11 = F.r10 * t01 + F.r11 * t11 + F.r12 * t21;
            float m12 = F.r10 * t02 + F.r11 * t12 + F.r12 * t22;
            float m20 = F.r20 * t00 + F.r21 * t10 + F.r22 * t20;
            float m21 = F.r20 * t01 + F.r21 * t11 + F.r22 * t21;
            float m22 = F.r20 * t02 + F.r21 * t12 + F.r22 * t22;
            float mp0 = F.r00 * tp0 + F.r01 * tp1 + F.r02 * tp2 + F.p0;
            float mp1 = F.r10 * tp0 + F.r11 * tp1 + F.r12 * tp2 + F.p1;
            float mp2 = F.r20 * tp0 + F.r21 * tp1 + F.r22 * tp2 + F.p2;
            float4* op = (float4*)(outs + b48 + (size_t)k * 16);
            op[0] = make_float4(m00, m01, m02, mp0);
            op[1] = make_float4(m10, m11, m12, mp1);
            op[2] = make_float4(m20, m21, m22, mp2);
            op[3] = make_float4(0.f, 0.f, 0.f, 1.f);
        }
    }
}

extern "C" void kernel_launch(void* const* d_in, const int* in_sizes, int n_in,
                              void* d_out, int out_size, void* d_ws, size_t ws_size,
                              hipStream_t stream) {
    const float* q   = (const float*)d_in[0];  // (B,12,2)
    const float* rev = (const float*)d_in[1];  // (12,6)
    const float* pri = (const float*)d_in[2];  // (12,6)
    const float* pos = (const float*)d_in[3];  // (12,3)
    const float* ori = (const float*)d_in[4];  // (12,3)

    const int B = in_sizes[0] / 24;
    float* out   = (float*)d_out;
    float* outs  = out;                        // B*48
    float* revls = out + (size_t)B * 48;       // B*72
    float* prils = out + (size_t)B * 120;      // B*72

    const int grid = (B + BLK - 1) / BLK;

    if (ws_size >= (size_t)CST_BYTES) {
        float* cst = (float*)d_ws;
        poe_precompute<<<1, 32, 0, stream>>>(rev, pri, pos, ori, cst);
        poe_fk_kernel<<<grid, BLK, 0, stream>>>(q, cst, outs, revls, prils, B);
    } else {
        poe_fk_fallback<<<grid, BLK, 0, stream>>>(q, rev, pri, pos, ori,
                                                  outs, revls, prils, B);
    }
}